// DynamicGaussians_13546326852097
// MI455X (gfx1250) — compile-verified
//
#include <hip/hip_runtime.h>
#include <hip/hip_bf16.h>

typedef float v2f __attribute__((ext_vector_type(2)));
typedef float v8f __attribute__((ext_vector_type(8)));

#define NGAUSS 1024
#define IMG_W 256
#define IMG_H 256
#define NPIX (IMG_W * IMG_H)

// ---- workspace layout (floats) ----
#define WS_COEFS 0
#define WS_COLS  (WS_COEFS + NGAUSS * 8)
#define WS_COEFU (WS_COLS  + NGAUSS * 4)
#define WS_COLU  (WS_COEFU + NGAUSS * 8)
#define WS_ZC    (WS_COLU  + NGAUSS * 4)

// Coefficients are pre-scaled by -log2(e) so the raster loop can use the raw
// base-2 v_exp_f32 directly:  s = -log2e * sigma;  alpha = opac * exp2(s).
#define NEG_LOG2E (-1.4426950408889634f)

// ---------------------------------------------------------------------------
// Kernel 1: per-gaussian preprocessing.
//   sigma(px,py) = k0*px^2 + k1*py^2 + k2*px*py + k3*px + k4*py + k5
// stored scaled by -log2(e), padded with k6=k7=0 for two K=4 WMMAs.
// ---------------------------------------------------------------------------
__global__ __launch_bounds__(256) void k_preprocess(
    const float* __restrict__ means, const float* __restrict__ scales,
    const float* __restrict__ quats, const float* __restrict__ opacities,
    const float* __restrict__ sh0,   const float* __restrict__ shn,
    const float* __restrict__ times, const float* __restrict__ durations,
    const float* __restrict__ velocities, const float* __restrict__ tptr,
    const float* __restrict__ w2c,   const float* __restrict__ intr,
    float* __restrict__ ws)
{
    int n = blockIdx.x * blockDim.x + threadIdx.x;
    if (n >= NGAUSS) return;

    float* coefU = ws + WS_COEFU;
    float* colU  = ws + WS_COLU;
    float* zcOut = ws + WS_ZC;

    const float t = tptr[0];
    float Rcw[3][3], tcw[3];
    #pragma unroll
    for (int i = 0; i < 3; ++i) {
        #pragma unroll
        for (int j = 0; j < 3; ++j) Rcw[i][j] = w2c[i * 4 + j];
        tcw[i] = w2c[i * 4 + 3];
    }
    const float fx = intr[0], fy = intr[4], cx = intr[2], cy = intr[5];

    // time-shifted mean + opacity
    float dt = t - times[n];
    float mx = means[n * 3 + 0] + dt * velocities[n * 3 + 0];
    float my = means[n * 3 + 1] + dt * velocities[n * 3 + 1];
    float mz = means[n * 3 + 2] + dt * velocities[n * 3 + 2];
    float op = 1.0f / (1.0f + expf(-opacities[n]));
    float qd = dt / expf(durations[n]);
    op *= expf(-0.5f * qd * qd);

    float sg[3] = { expf(scales[n * 3 + 0]), expf(scales[n * 3 + 1]), expf(scales[n * 3 + 2]) };

    // camera-space position
    float pc[3];
    #pragma unroll
    for (int i = 0; i < 3; ++i)
        pc[i] = Rcw[i][0] * mx + Rcw[i][1] * my + Rcw[i][2] * mz + tcw[i];
    float zc = pc[2];
    float iz = 1.0f / zc;
    float u = fx * pc[0] * iz + cx;
    float v = fy * pc[1] * iz + cy;

    // quaternion -> rotation
    float qw = quats[n * 4 + 0], qx = quats[n * 4 + 1], qy = quats[n * 4 + 2], qz = quats[n * 4 + 3];
    float qn = rsqrtf(qw * qw + qx * qx + qy * qy + qz * qz);
    qw *= qn; qx *= qn; qy *= qn; qz *= qn;
    float R[3][3] = {
        {1.f - 2.f * (qy * qy + qz * qz), 2.f * (qx * qy - qw * qz),       2.f * (qx * qz + qw * qy)},
        {2.f * (qx * qy + qw * qz),       1.f - 2.f * (qx * qx + qz * qz), 2.f * (qy * qz - qw * qx)},
        {2.f * (qx * qz - qw * qy),       2.f * (qy * qz + qw * qx),       1.f - 2.f * (qx * qx + qy * qy)}
    };

    // cov3d = (R * diag(sg)) (R * diag(sg))^T
    float Mm[3][3];
    #pragma unroll
    for (int i = 0; i < 3; ++i)
        #pragma unroll
        for (int j = 0; j < 3; ++j) Mm[i][j] = R[i][j] * sg[j];
    float cov3[3][3];
    #pragma unroll
    for (int i = 0; i < 3; ++i)
        #pragma unroll
        for (int k = 0; k < 3; ++k)
            cov3[i][k] = Mm[i][0] * Mm[k][0] + Mm[i][1] * Mm[k][1] + Mm[i][2] * Mm[k][2];

    // cov_cam = Rcw * cov3 * Rcw^T
    float TT[3][3];
    #pragma unroll
    for (int i = 0; i < 3; ++i)
        #pragma unroll
        for (int k = 0; k < 3; ++k)
            TT[i][k] = Rcw[i][0] * cov3[0][k] + Rcw[i][1] * cov3[1][k] + Rcw[i][2] * cov3[2][k];
    float cc[3][3];
    #pragma unroll
    for (int i = 0; i < 3; ++i)
        #pragma unroll
        for (int l = 0; l < 3; ++l)
            cc[i][l] = TT[i][0] * Rcw[l][0] + TT[i][1] * Rcw[l][1] + TT[i][2] * Rcw[l][2];

    // cov2d = J * cov_cam * J^T
    float J0[3] = { fx * iz, 0.f, -fx * pc[0] * iz * iz };
    float J1[3] = { 0.f, fy * iz, -fy * pc[1] * iz * iz };
    float JC0[3], JC1[3];
    #pragma unroll
    for (int k = 0; k < 3; ++k) {
        JC0[k] = J0[0] * cc[0][k] + J0[1] * cc[1][k] + J0[2] * cc[2][k];
        JC1[k] = J1[0] * cc[0][k] + J1[1] * cc[1][k] + J1[2] * cc[2][k];
    }
    float a = JC0[0] * J0[0] + JC0[1] * J0[1] + JC0[2] * J0[2] + 0.3f;
    float b = JC0[0] * J1[0] + JC0[1] * J1[1] + JC0[2] * J1[2];
    float c = JC1[0] * J1[0] + JC1[1] * J1[1] + JC1[2] * J1[2] + 0.3f;
    float det = a * c - b * b;
    bool valid = (zc > 0.01f) && (det > 1e-12f);
    float idet = 1.0f / det;
    float conA = c * idet, conB = -b * idet, conC = a * idet;

    // view direction from camera position
    float cpx = -(Rcw[0][0] * tcw[0] + Rcw[1][0] * tcw[1] + Rcw[2][0] * tcw[2]);
    float cpy = -(Rcw[0][1] * tcw[0] + Rcw[1][1] * tcw[1] + Rcw[2][1] * tcw[2]);
    float cpz = -(Rcw[0][2] * tcw[0] + Rcw[1][2] * tcw[1] + Rcw[2][2] * tcw[2]);
    float dx = mx - cpx, dy = my - cpy, dz = mz - cpz;
    float il = rsqrtf(dx * dx + dy * dy + dz * dz);
    float x = dx * il, y = dy * il, z = dz * il;

    // SH basis (degree 3, 16 terms)
    float xx = x * x, yy = y * y, zz = z * z;
    float basis[16];
    basis[0]  = 0.28209479177387814f;
    basis[1]  = -0.4886025119029199f * y;
    basis[2]  =  0.4886025119029199f * z;
    basis[3]  = -0.4886025119029199f * x;
    basis[4]  =  1.0925484305920792f * x * y;
    basis[5]  = -1.0925484305920792f * y * z;
    basis[6]  =  0.31539156525252005f * (2.f * zz - xx - yy);
    basis[7]  = -1.0925484305920792f * x * z;
    basis[8]  =  0.5462742152960396f * (xx - yy);
    basis[9]  = -0.5900435899266435f * y * (3.f * xx - yy);
    basis[10] =  2.890611442640554f  * x * y * z;
    basis[11] = -0.4570457994644658f * y * (4.f * zz - xx - yy);
    basis[12] =  0.3731763325901154f * z * (2.f * zz - 3.f * xx - 3.f * yy);
    basis[13] = -0.4570457994644658f * x * (4.f * zz - xx - yy);
    basis[14] =  1.445305721320277f  * z * (xx - yy);
    basis[15] = -0.5900435899266435f * x * (xx - 3.f * yy);

    float col[3];
    #pragma unroll
    for (int ch = 0; ch < 3; ++ch) {
        float s = basis[0] * sh0[n * 3 + ch];
        #pragma unroll
        for (int k = 1; k < 16; ++k)
            s += basis[k] * shn[(n * 15 + (k - 1)) * 3 + ch];
        col[ch] = fmaxf(s + 0.5f, 0.0f);
    }

    // quadratic-form coefficients, pre-scaled by -log2(e)
    float a2 = 0.5f * conA, c2 = 0.5f * conC;
    coefU[n * 8 + 0] = NEG_LOG2E * a2;
    coefU[n * 8 + 1] = NEG_LOG2E * c2;
    coefU[n * 8 + 2] = NEG_LOG2E * conB;
    coefU[n * 8 + 3] = NEG_LOG2E * (-(conA * u + conB * v));
    coefU[n * 8 + 4] = NEG_LOG2E * (-(conC * v + conB * u));
    coefU[n * 8 + 5] = NEG_LOG2E * (a2 * u * u + c2 * v * v + conB * u * v);
    coefU[n * 8 + 6] = 0.f;
    coefU[n * 8 + 7] = 0.f;

    colU[n * 4 + 0] = col[0];
    colU[n * 4 + 1] = col[1];
    colU[n * 4 + 2] = col[2];
    colU[n * 4 + 3] = valid ? op : 0.0f;
    zcOut[n] = zc;
}

// ---------------------------------------------------------------------------
// Kernel 2: stable rank sort by depth + scatter into sorted arrays.
// ---------------------------------------------------------------------------
__global__ __launch_bounds__(256) void k_sort_scatter(float* __restrict__ ws)
{
    int i = blockIdx.x * blockDim.x + threadIdx.x;
    if (i >= NGAUSS) return;
    const float* zc    = ws + WS_ZC;
    const float* coefU = ws + WS_COEFU;
    const float* colU  = ws + WS_COLU;
    float* coefS = ws + WS_COEFS;
    float* colS  = ws + WS_COLS;

    float z = zc[i];
    int rank = 0;
    for (int j = 0; j < NGAUSS; ++j) {
        float zj = zc[j];
        rank += (zj < z) || (zj == z && j < i);   // stable, matches argsort
    }
    #pragma unroll
    for (int k = 0; k < 8; ++k) coefS[rank * 8 + k] = coefU[i * 8 + k];
    #pragma unroll
    for (int k = 0; k < 4; ++k) colS[rank * 4 + k] = colU[i * 4 + k];
}

// ---------------------------------------------------------------------------
// Kernel 3: WMMA rasterizer.
// Block = 256 threads = 8 waves; each wave owns 16 pixels. All sorted gaussian
// data staged to LDS once (48 KB of 320 KB/WGP). Per 16-gaussian tile:
// sigma tile via 2x v_wmma_f32_16x16x4_f32, branchless alpha with raw
// v_exp_f32 (__builtin_amdgcn_exp2f), associative composite via shfl_xor(16).
// ---------------------------------------------------------------------------
__global__ __launch_bounds__(256) void k_raster(const float* __restrict__ ws,
                                                float* __restrict__ out)
{
    __shared__ float sCoef[NGAUSS * 8];   // 32 KB
    __shared__ float sCol [NGAUSS * 4];   // 16 KB

    const float* coefS = ws + WS_COEFS;
    const float* colS  = ws + WS_COLS;

    int tid = threadIdx.x;
    for (int g = tid; g < NGAUSS; g += 256) {
        float4 c0 = ((const float4*)coefS)[g * 2 + 0];
        float4 c1 = ((const float4*)coefS)[g * 2 + 1];
        ((float4*)sCoef)[g * 2 + 0] = c0;
        ((float4*)sCoef)[g * 2 + 1] = c1;
        ((float4*)sCol)[g] = ((const float4*)colS)[g];
    }
    __syncthreads();

    int lane  = tid & 31;
    int wave  = tid >> 5;
    int col16 = lane & 15;        // pixel column within tile / gaussian row owner
    int half  = lane >> 4;        // 0: rows 0..7, 1: rows 8..15 (C/D layout)

    int p  = blockIdx.x * 128 + wave * 16 + col16;   // global pixel id
    float px = (float)(p & (IMG_W - 1)) + 0.5f;
    float py = (float)(p >> 8) + 0.5f;

    // B operand: pixel basis [px^2, py^2, px*py, px, py, 1, 0, 0]
    float basis[8] = { px * px, py * py, px * py, px, py, 1.0f, 0.0f, 0.0f };
    v2f bLo, bHi;
    bLo[0] = basis[half * 2 + 0];  bLo[1] = basis[half * 2 + 1];
    bHi[0] = basis[4 + half * 2];  bHi[1] = basis[5 + half * 2];

    float Trun = 1.0f, rR = 0.0f, rG = 0.0f, rB = 0.0f;

    for (int tIdx = 0; tIdx < NGAUSS / 16; ++tIdx) {
        // A operand: 16 gaussians' coefficients (depth-sorted)
        int g = tIdx * 16 + col16;
        const float* cf = &sCoef[g * 8];
        v2f aLo, aHi;
        aLo[0] = cf[half * 2 + 0];  aLo[1] = cf[half * 2 + 1];
        aHi[0] = cf[4 + half * 2];  aHi[1] = cf[5 + half * 2];

        v8f acc = {};
        acc = __builtin_amdgcn_wmma_f32_16x16x4_f32(false, aLo, false, bLo,
                                                    (short)0, acc, false, false);
        acc = __builtin_amdgcn_wmma_f32_16x16x4_f32(false, aHi, false, bHi,
                                                    (short)0, acc, false, false);

        // local composite of this half's 8 gaussian rows, fully branchless.
        // acc[r] = s = -log2e * sigma; sigma>=0 <=> s<=0.
        // Raw v_exp_f32: s>0 overflows but is gated off; s<-126 flushes to 0,
        // which is already below the 1/255 alpha cut. No range fixup needed.
        float Tl = 1.0f, lR = 0.0f, lG = 0.0f, lB = 0.0f;
        #pragma unroll
        for (int r = 0; r < 8; ++r) {
            int gr = tIdx * 16 + half * 8 + r;
            float4 cl = ((const float4*)sCol)[gr];   // r,g,b,opac (LDS broadcast)
            float s = acc[r];
            float e = __builtin_amdgcn_exp2f(s);      // raw v_exp_f32
            float alpha = fminf(cl.w * e, 0.999f);
            alpha = ((s <= 0.0f) && (alpha >= (1.0f / 255.0f))) ? alpha : 0.0f;
            float wgt = Tl * alpha;
            lR = fmaf(wgt, cl.x, lR);
            lG = fmaf(wgt, cl.y, lG);
            lB = fmaf(wgt, cl.z, lB);
            Tl = fmaf(-alpha, Tl, Tl);
        }

        // merge halves: front = rows 0..7 (half 0), back = rows 8..15 (half 1)
        float oR = __shfl_xor(lR, 16, 32);
        float oG = __shfl_xor(lG, 16, 32);
        float oB = __shfl_xor(lB, 16, 32);
        float oT = __shfl_xor(Tl, 16, 32);
        float fR = half ? oR : lR, fG = half ? oG : lG, fB = half ? oB : lB, fT = half ? oT : Tl;
        float bR = half ? lR : oR, bG = half ? lG : oG, bB = half ? lB : oB, bT = half ? Tl : oT;
        float tR = fmaf(fT, bR, fR);
        float tG = fmaf(fT, bG, fG);
        float tB = fmaf(fT, bB, fB);
        float tT = fT * bT;

        rR = fmaf(Trun, tR, rR);
        rG = fmaf(Trun, tG, rG);
        rB = fmaf(Trun, tB, rB);
        Trun *= tT;
    }

    if (half == 0) {
        out[p * 3 + 0] = fminf(fmaxf(rR, 0.0f), 1.0f);
        out[p * 3 + 1] = fminf(fmaxf(rG, 0.0f), 1.0f);
        out[p * 3 + 2] = fminf(fmaxf(rB, 0.0f), 1.0f);
        out[NPIX * 3 + p] = 1.0f - Trun;
    }
}

// ---------------------------------------------------------------------------
extern "C" void kernel_launch(void* const* d_in, const int* in_sizes, int n_in,
                              void* d_out, int out_size, void* d_ws, size_t ws_size,
                              hipStream_t stream)
{
    const float* means      = (const float*)d_in[0];
    const float* scales     = (const float*)d_in[1];
    const float* quats      = (const float*)d_in[2];
    const float* opacities  = (const float*)d_in[3];
    const float* sh0        = (const float*)d_in[4];
    const float* shn        = (const float*)d_in[5];
    const float* times      = (const float*)d_in[6];
    const float* durations  = (const float*)d_in[7];
    const float* velocities = (const float*)d_in[8];
    const float* tptr       = (const float*)d_in[9];
    const float* w2c        = (const float*)d_in[10];
    const float* intr       = (const float*)d_in[11];
    float* ws  = (float*)d_ws;
    float* out = (float*)d_out;

    k_preprocess<<<NGAUSS / 256, 256, 0, stream>>>(means, scales, quats, opacities,
                                                   sh0, shn, times, durations,
                                                   velocities, tptr, w2c, intr, ws);
    k_sort_scatter<<<NGAUSS / 256, 256, 0, stream>>>(ws);
    k_raster<<<NPIX / 128, 256, 0, stream>>>(ws, out);
}